// TopKPool_32392643346828
// MI455X (gfx1250) — compile-verified
//
#include <hip/hip_runtime.h>
#include <math.h>

typedef __attribute__((ext_vector_type(2))) float v2f;
typedef __attribute__((ext_vector_type(8))) float v8f;

#define HDIM 128
#define NB 100        // graphs
#define LDSPITCH 132  // padded LDS row stride (floats): 132 mod 64 = 4 -> no bank conflicts

// ---------------------------------------------------------------------------
// GEMM: Hout[M,128] = X[M,128] @ W[128,128]   (fp32 WMMA 16x16x4)
// One wave computes a 16x16 tile; 8 waves/block cover 16 rows x 128 cols.
// A tile staged into LDS with CDNA5 async global->LDS copies; B fragment
// preloaded into registers so the WMMA chain runs back-to-back.
// NOTE: the LDS operand of the async copy is derived from the pointer
// ((unsigned)(size_t)&As[...]) so the array is captured and the compiler
// cannot fold the later ds_loads as reads of never-written memory. The low
// 32 bits of a flat LDS address are the workgroup-relative LDS byte offset
// (ISA 10.2), which is exactly what the instruction's VDST expects.
// ---------------------------------------------------------------------------
__global__ void gemm_xw_wmma(const float* __restrict__ X,
                             const float* __restrict__ W,
                             float* __restrict__ Hout, int M) {
  __shared__ float As[16 * LDSPITCH];
  const int t = threadIdx.x;
  const int lane = t & 31;
  const int wave = t >> 5;           // 0..7 -> column tile
  const int rowBase = blockIdx.x * 16;

  // --- async-stage 16x128 A tile into LDS (rows clamped for tail tiles) ---
#pragma unroll
  for (int it = 0; it < 2; ++it) {
    int idx = t + it * 256;          // 0..511  (16 rows x 32 float4)
    int row = idx >> 5;              // 0..15
    int c4 = (idx & 31) * 4;         // 0,4,...,124
    int grow = rowBase + row;
    if (grow >= M) grow = M - 1;     // duplicate row; out-of-range D rows not stored
    const float* gp = X + (size_t)grow * HDIM + c4;
    float* lp = &As[row * LDSPITCH + c4];
    asm volatile("global_load_async_to_lds_b128 %0, %1, off"
                 :
                 : "v"((unsigned)(size_t)lp),
                   "v"((unsigned long long)(size_t)gp)
                 : "memory");
  }
  asm volatile("s_wait_asynccnt 0x0" ::: "memory");
  __syncthreads();

  const int colBase = wave * 16;
  const int r = lane & 15;
  const int kb = (lane >> 4) * 2;    // 0 or 2
  const int col = colBase + r;

  // --- preload B fragment (64 floats per lane) into registers ---
  float breg[64];
#pragma unroll
  for (int s = 0; s < 32; ++s) {
    int k0 = s * 4;
    breg[2 * s]     = W[(k0 + kb) * HDIM + col];
    breg[2 * s + 1] = W[(k0 + kb + 1) * HDIM + col];
  }

  const float* arow = &As[r * LDSPITCH];
  v8f acc = {};
#pragma unroll
  for (int s = 0; s < 32; ++s) {
    int k0 = s * 4;
    v2f a = *(const v2f*)(arow + k0 + kb);  // ds_load_b64, conflict-free
    v2f b;
    b.x = breg[2 * s];
    b.y = breg[2 * s + 1];
    acc = __builtin_amdgcn_wmma_f32_16x16x4_f32(
        /*neg_a=*/false, a, /*neg_b=*/false, b,
        /*c_mod=*/(short)0, acc, /*reuse_a=*/false, /*reuse_b=*/false);
  }

  const int mOff = (lane >> 4) * 8;
#pragma unroll
  for (int v = 0; v < 8; ++v) {
    int m = rowBase + mOff + v;
    if (m < M) Hout[(size_t)m * HDIM + col] = acc[v];
  }
}

// ---------------------------------------------------------------------------
// GCN conv helpers
// ---------------------------------------------------------------------------
__global__ void fill_deg_one(float* deg, int M) {
  int i = blockIdx.x * blockDim.x + threadIdx.x;
  if (i < M) deg[i] = 1.0f;
}

__global__ void edge_deg(const int* __restrict__ edst,
                         const float* __restrict__ ew, float* deg, int E) {
  int e = blockIdx.x * blockDim.x + threadIdx.x;
  if (e < E) {
    float w = ew[e];
    if (w != 0.0f) atomicAdd(&deg[edst[e]], w);
  }
}

__global__ void compute_dinv(const float* deg, float* dinv, int M) {
  int i = blockIdx.x * blockDim.x + threadIdx.x;
  if (i < M) dinv[i] = rsqrtf(deg[i]);
}

__global__ void edge_coef(const int* __restrict__ esrc,
                          const int* __restrict__ edst,
                          const float* __restrict__ ew,
                          const float* __restrict__ dinv,
                          float* __restrict__ coef, int E) {
  int e = blockIdx.x * blockDim.x + threadIdx.x;
  if (e < E) coef[e] = dinv[esrc[e]] * dinv[edst[e]] * ew[e];
}

// acc[i,f] = dinv[i]^2 * h[i,f] + b[f]
__global__ void init_acc(const float* __restrict__ h,
                         const float* __restrict__ dinv,
                         const float* __restrict__ bias,
                         float* __restrict__ acc, int M) {
  long t = (long)blockIdx.x * blockDim.x + threadIdx.x;
  if (t < (long)M * HDIM) {
    int i = (int)(t >> 7);
    int f = (int)(t & (HDIM - 1));
    float di = dinv[i];
    acc[t] = di * di * h[t] + bias[f];
  }
}

// one wave per edge; each lane scatters 4 features with float atomics
__global__ void edge_scatter(const float* __restrict__ h,
                             const int* __restrict__ esrc,
                             const int* __restrict__ edst,
                             const float* __restrict__ coef,
                             float* __restrict__ acc, int E) {
  long tid = (long)blockIdx.x * blockDim.x + threadIdx.x;
  int e = (int)(tid >> 5);
  if (e >= E) return;
  float c = coef[e];
  if (c == 0.0f) return;
  int lane = (int)(tid & 31);
  int s = esrc[e], d = edst[e];
  const float4 v = ((const float4*)(h + (size_t)s * HDIM))[lane];
  float* ad = acc + (size_t)d * HDIM + lane * 4;
  atomicAdd(ad + 0, c * v.x);
  atomicAdd(ad + 1, c * v.y);
  atomicAdd(ad + 2, c * v.z);
  atomicAdd(ad + 3, c * v.w);
}

__global__ void tanh_inplace(float* x, long n) {
  long t = (long)blockIdx.x * blockDim.x + threadIdx.x;
  if (t < n) x[t] = tanhf(x[t]);
}

// ---------------------------------------------------------------------------
// Top-K pooling
// ---------------------------------------------------------------------------
__global__ void pnorm_kernel(const float* __restrict__ p, float* invn) {
  __shared__ float red[HDIM];
  int t = threadIdx.x;
  float v = p[t];
  red[t] = v * v;
  __syncthreads();
  for (int s = HDIM / 2; s > 0; s >>= 1) {
    if (t < s) red[t] += red[t + s];
    __syncthreads();
  }
  if (t == 0) invn[0] = rsqrtf(red[0]);
}

__global__ void score_kernel(const float* __restrict__ x,
                             const float* __restrict__ p,
                             const float* __restrict__ invn,
                             float* __restrict__ score, int M) {
  __shared__ float red[HDIM];
  int i = blockIdx.x, t = threadIdx.x;
  red[t] = x[(size_t)i * HDIM + t] * p[t];
  __syncthreads();
  for (int s = HDIM / 2; s > 0; s >>= 1) {
    if (t < s) red[t] += red[t + s];
    __syncthreads();
  }
  if (t == 0) score[i] = red[0] * invn[0];
}

__global__ void reset_newid(int* newid, int M) {
  int i = blockIdx.x * blockDim.x + threadIdx.x;
  if (i < M) newid[i] = -1;
}

// bitonic sort of 512 (score,idx) pairs, descending; keep top k per graph
__global__ void topk_sort(const float* __restrict__ score, int n_per, int k,
                          int* __restrict__ permIdx,
                          float* __restrict__ permScore) {
  __shared__ float s[512];
  __shared__ int si[512];
  int g = blockIdx.x, t = threadIdx.x;
  int base = g * n_per;
  s[t] = (t < n_per) ? score[base + t] : -INFINITY;
  si[t] = t;
  __syncthreads();
  for (int size = 2; size <= 512; size <<= 1) {
    for (int stride = size >> 1; stride > 0; stride >>= 1) {
      int partner = t ^ stride;
      if (partner > t) {
        bool desc = ((t & size) == 0);
        float a = s[t], b = s[partner];
        bool doSwap = desc ? (a < b) : (a > b);
        if (doSwap) {
          s[t] = b; s[partner] = a;
          int ti = si[t]; si[t] = si[partner]; si[partner] = ti;
        }
      }
      __syncthreads();
    }
  }
  if (t < k) {
    permIdx[g * k + t] = base + si[t];
    permScore[g * k + t] = s[t];
  }
}

// xk[newRow] = x[oldRow] * tanh(score); newid[oldRow] = newRow
__global__ void topk_gather(const float* __restrict__ x,
                            const int* __restrict__ permIdx,
                            const float* __restrict__ permScore,
                            float* __restrict__ xk, int* __restrict__ newid) {
  int nr = blockIdx.x, t = threadIdx.x;
  int oldRow = permIdx[nr];
  float gate = tanhf(permScore[nr]);
  xk[(size_t)nr * HDIM + t] = x[(size_t)oldRow * HDIM + t] * gate;
  if (t == 0) newid[oldRow] = nr;
}

__global__ void edge_relabel(int* esrc, int* edst, float* ew,
                             const int* __restrict__ newid, int E) {
  int e = blockIdx.x * blockDim.x + threadIdx.x;
  if (e >= E) return;
  int ns = newid[esrc[e]];
  int nd = newid[edst[e]];
  bool valid = (ns >= 0) && (nd >= 0) && (ew[e] > 0.0f);
  esrc[e] = valid ? ns : 0;
  edst[e] = valid ? nd : 0;
  ew[e] = valid ? ew[e] : 0.0f;
}

__global__ void init_edges(const int* __restrict__ src0,
                           const int* __restrict__ dst0,
                           int* esrc, int* edst, float* ew, int E) {
  int e = blockIdx.x * blockDim.x + threadIdx.x;
  if (e < E) { esrc[e] = src0[e]; edst[e] = dst0[e]; ew[e] = 1.0f; }
}

// ---------------------------------------------------------------------------
// Readout + classifier
// ---------------------------------------------------------------------------
__global__ void readout_kernel(const float* __restrict__ xk, int k,
                               float* __restrict__ gr, int accumulate) {
  int g = blockIdx.x, t = threadIdx.x;  // 128 threads, 1 feature each
  const float* base = xk + (size_t)g * k * HDIM;
  float sum = 0.0f, mx = -INFINITY;
  for (int i = 0; i < k; ++i) {
    float v = base[(size_t)i * HDIM + t];
    sum += v;
    mx = fmaxf(mx, v);
  }
  float mean = sum / (float)k;
  float* go = gr + (size_t)g * 3 * HDIM;
  if (accumulate) {
    go[t] += mean; go[HDIM + t] += mx; go[2 * HDIM + t] += sum;
  } else {
    go[t] = mean; go[HDIM + t] = mx; go[2 * HDIM + t] = sum;
  }
}

__global__ void mlp_kernel(const float* __restrict__ gr,
                           const float* __restrict__ lw1,
                           const float* __restrict__ lb1,
                           const float* __restrict__ lw2,
                           const float* __restrict__ lb2,
                           float* __restrict__ out) {
  __shared__ float gin[3 * HDIM];
  __shared__ float h1[HDIM];
  __shared__ float logits[6];
  int g = blockIdx.x, t = threadIdx.x;  // 128 threads
  for (int i = t; i < 3 * HDIM; i += HDIM) gin[i] = gr[(size_t)g * 3 * HDIM + i];
  __syncthreads();
  float a = lb1[t];
  for (int i = 0; i < 3 * HDIM; ++i) a += gin[i] * lw1[i * HDIM + t];
  h1[t] = fmaxf(a, 0.0f);
  __syncthreads();
  if (t < 6) {
    float c = lb2[t];
    for (int j = 0; j < HDIM; ++j) c += h1[j] * lw2[j * 6 + t];
    logits[t] = c;
  }
  __syncthreads();
  if (t == 0) {
    float mx = logits[0];
    for (int c = 1; c < 6; ++c) mx = fmaxf(mx, logits[c]);
    float se = 0.0f;
    for (int c = 0; c < 6; ++c) se += expf(logits[c] - mx);
    float lse = mx + logf(se);
    for (int c = 0; c < 6; ++c) out[g * 6 + c] = logits[c] - lse;
  }
}

// ---------------------------------------------------------------------------
// Host-side orchestration
// ---------------------------------------------------------------------------
static inline int cdiv(long a, long b) { return (int)((a + b - 1) / b); }

struct ConvBufs {
  float *hbuf, *deg, *dinv, *coef;
  int *esrc, *edst;
  float *ew;
};

static void run_conv(const float* x, const float* W, const float* bias,
                     float* acc, int M, int E, const ConvBufs& cb,
                     hipStream_t stream) {
  gemm_xw_wmma<<<cdiv(M, 16), 256, 0, stream>>>(x, W, cb.hbuf, M);
  fill_deg_one<<<cdiv(M, 256), 256, 0, stream>>>(cb.deg, M);
  edge_deg<<<cdiv(E, 256), 256, 0, stream>>>(cb.edst, cb.ew, cb.deg, E);
  compute_dinv<<<cdiv(M, 256), 256, 0, stream>>>(cb.deg, cb.dinv, M);
  edge_coef<<<cdiv(E, 256), 256, 0, stream>>>(cb.esrc, cb.edst, cb.ew, cb.dinv,
                                              cb.coef, E);
  init_acc<<<cdiv((long)M * HDIM, 256), 256, 0, stream>>>(cb.hbuf, cb.dinv,
                                                          bias, acc, M);
  edge_scatter<<<cdiv((long)E * 32, 256), 256, 0, stream>>>(
      cb.hbuf, cb.esrc, cb.edst, cb.coef, acc, E);
  tanh_inplace<<<cdiv((long)M * HDIM, 256), 256, 0, stream>>>(acc,
                                                              (long)M * HDIM);
}

extern "C" void kernel_launch(void* const* d_in, const int* in_sizes, int n_in,
                              void* d_out, int out_size, void* d_ws,
                              size_t ws_size, hipStream_t stream) {
  const float* x0  = (const float*)d_in[0];
  const int*   s0  = (const int*)d_in[1];
  const int*   d0  = (const int*)d_in[2];
  const float* W1  = (const float*)d_in[3];
  const float* b1  = (const float*)d_in[4];
  const float* W2  = (const float*)d_in[5];
  const float* b2  = (const float*)d_in[6];
  const float* W3  = (const float*)d_in[7];
  const float* b3  = (const float*)d_in[8];
  const float* W4  = (const float*)d_in[9];
  const float* b4  = (const float*)d_in[10];
  const float* p1  = (const float*)d_in[11];
  const float* p2  = (const float*)d_in[12];
  const float* p3  = (const float*)d_in[13];
  const float* lw1 = (const float*)d_in[14];
  const float* lb1 = (const float*)d_in[15];
  const float* lw2 = (const float*)d_in[16];
  const float* lb2 = (const float*)d_in[17];
  float* out = (float*)d_out;

  const int N0 = in_sizes[0] / HDIM;  // 50000
  const int E  = in_sizes[1];         // 500000
  const int nper0 = N0 / NB;          // 500

  // --- workspace carve-out (256B aligned) ---
  char* base = (char*)d_ws;
  size_t off = 0;
  auto take = [&](size_t bytes) -> void* {
    off = (off + 255) & ~(size_t)255;
    void* p = base + off;
    off += bytes;
    return p;
  };
  float* bufA  = (float*)take((size_t)N0 * HDIM * sizeof(float));
  float* bufB  = (float*)take((size_t)N0 * HDIM * sizeof(float));
  float* hbuf  = (float*)take((size_t)N0 * HDIM * sizeof(float));
  float* deg   = (float*)take((size_t)N0 * sizeof(float));
  float* dinv  = (float*)take((size_t)N0 * sizeof(float));
  float* score = (float*)take((size_t)N0 * sizeof(float));
  float* coef  = (float*)take((size_t)E * sizeof(float));
  int*   esrc  = (int*)take((size_t)E * sizeof(int));
  int*   edst  = (int*)take((size_t)E * sizeof(int));
  float* ew    = (float*)take((size_t)E * sizeof(float));
  int*   newid = (int*)take((size_t)N0 * sizeof(int));
  int*   permI = (int*)take((size_t)NB * 250 * sizeof(int));
  float* permS = (float*)take((size_t)NB * 250 * sizeof(float));
  float* gr    = (float*)take((size_t)NB * 3 * HDIM * sizeof(float));
  float* invn  = (float*)take(sizeof(float));
  (void)ws_size; (void)n_in; (void)out_size;

  ConvBufs cb{hbuf, deg, dinv, coef, esrc, edst, ew};

  init_edges<<<cdiv(E, 256), 256, 0, stream>>>(s0, d0, esrc, edst, ew, E);

  // conv1: x0 -> bufA ; conv2: bufA -> bufB
  run_conv(x0, W1, b1, bufA, N0, E, cb, stream);
  run_conv(bufA, W2, b2, bufB, N0, E, cb, stream);

  // pool1: bufB (N0, nper0=500) -> bufA (N1, k1=250)
  int k1 = (nper0 + 1) / 2, N1 = NB * k1;
  pnorm_kernel<<<1, HDIM, 0, stream>>>(p1, invn);
  score_kernel<<<N0, HDIM, 0, stream>>>(bufB, p1, invn, score, N0);
  reset_newid<<<cdiv(N0, 256), 256, 0, stream>>>(newid, N0);
  topk_sort<<<NB, 512, 0, stream>>>(score, nper0, k1, permI, permS);
  topk_gather<<<N1, HDIM, 0, stream>>>(bufB, permI, permS, bufA, newid);
  edge_relabel<<<cdiv(E, 256), 256, 0, stream>>>(esrc, edst, ew, newid, E);
  readout_kernel<<<NB, HDIM, 0, stream>>>(bufA, k1, gr, 0);

  // conv3: bufA -> bufB
  run_conv(bufA, W3, b3, bufB, N1, E, cb, stream);

  // pool2: bufB (N1, k1) -> bufA (N2, k2=125)
  int k2 = (k1 + 1) / 2, N2 = NB * k2;
  pnorm_kernel<<<1, HDIM, 0, stream>>>(p2, invn);
  score_kernel<<<N1, HDIM, 0, stream>>>(bufB, p2, invn, score, N1);
  reset_newid<<<cdiv(N1, 256), 256, 0, stream>>>(newid, N1);
  topk_sort<<<NB, 512, 0, stream>>>(score, k1, k2, permI, permS);
  topk_gather<<<N2, HDIM, 0, stream>>>(bufB, permI, permS, bufA, newid);
  edge_relabel<<<cdiv(E, 256), 256, 0, stream>>>(esrc, edst, ew, newid, E);
  readout_kernel<<<NB, HDIM, 0, stream>>>(bufA, k2, gr, 1);

  // conv4: bufA -> bufB
  run_conv(bufA, W4, b4, bufB, N2, E, cb, stream);

  // pool3: bufB (N2, k2) -> bufA (N3, k3=63)
  int k3 = (k2 + 1) / 2, N3 = NB * k3;
  pnorm_kernel<<<1, HDIM, 0, stream>>>(p3, invn);
  score_kernel<<<N2, HDIM, 0, stream>>>(bufB, p3, invn, score, N2);
  reset_newid<<<cdiv(N2, 256), 256, 0, stream>>>(newid, N2);
  topk_sort<<<NB, 512, 0, stream>>>(score, k2, k3, permI, permS);
  topk_gather<<<N3, HDIM, 0, stream>>>(bufB, permI, permS, bufA, newid);
  readout_kernel<<<NB, HDIM, 0, stream>>>(bufA, k3, gr, 1);

  // classifier head
  mlp_kernel<<<NB, HDIM, 0, stream>>>(gr, lw1, lb1, lw2, lb2, out);
}